// GCN_36773509988954
// MI455X (gfx1250) — compile-verified
//
#include <hip/hip_runtime.h>

#define DIM 128
#define LDA 132            // padded LDS stride for the A tile (kills 16-way bank conflicts)
#define TILES_PER_BLOCK 8  // 128 rows of output per block; W staged once and reused 8x

typedef __attribute__((ext_vector_type(2))) float v2f;
typedef __attribute__((ext_vector_type(8))) float v8f;

// ---------------------------------------------------------------- degree init
__global__ void init_deg_kernel(float* __restrict__ deg, int n) {
    int i = blockIdx.x * blockDim.x + threadIdx.x;
    if (i < n) deg[i] = 1.0f;  // self-loop contributes weight 1
}

__global__ void deg_accum_kernel(float* __restrict__ deg,
                                 const long long* __restrict__ ei,
                                 const float* __restrict__ ew, int E) {
    int e = blockIdx.x * blockDim.x + threadIdx.x;
    if (e < E) {
        long long c = ei[(long long)E + e];  // target node
        atomicAdd(&deg[c], ew[e]);
    }
}

__global__ void dinv_kernel(float* __restrict__ deg, int n) {
    int i = blockIdx.x * blockDim.x + threadIdx.x;
    if (i < n) {
        float d = deg[i];
        deg[i] = (d > 0.0f) ? rsqrtf(d) : 0.0f;  // in-place deg -> dinv
    }
}

__global__ void norm_kernel(float* __restrict__ norm,
                            const float* __restrict__ dinv,
                            const long long* __restrict__ ei,
                            const float* __restrict__ ew, int E) {
    int e = blockIdx.x * blockDim.x + threadIdx.x;
    if (e < E) {
        long long r = ei[e];
        long long c = ei[(long long)E + e];
        norm[e] = dinv[r] * ew[e] * dinv[c];
    }
}

// ------------------------------------------------------- GEMM: [n,128]@[128,128]
// 256 threads = 8 wave32s. Block stages W (64 KB) once, then sweeps 8 M-tiles
// with a double-buffered A tile so staging overlaps the WMMA loop.
// Per 16x16 tile: K=128 -> 32 steps of V_WMMA_F32_16X16X4_F32.
__global__ __launch_bounds__(256) void gemm128_wmma_kernel(
        const float* __restrict__ A, const float* __restrict__ W,
        float* __restrict__ Out, int nrows) {
    __shared__ float lds_w[DIM * DIM];       // 64 KB: whole weight matrix
    __shared__ float lds_a[2][16 * LDA];     // 2 x 8.25 KB: double-buffered A tile

    const int tid      = threadIdx.x;
    const int rowBase0 = blockIdx.x * (16 * TILES_PER_BLOCK);

    // Stage W (coalesced float4); W stays resident for all 8 tiles.
    const float4* Wv = (const float4*)W;
    float4* lwv = (float4*)lds_w;
    #pragma unroll
    for (int i = tid; i < DIM * DIM / 4; i += 256) lwv[i] = Wv[i];

    // Stage A tile 0: 16 rows x 32 float4 (zeros past nrows)
    {
        for (int i = tid; i < 16 * 32; i += 256) {
            int row = i >> 5;
            int cv  = i & 31;
            float4 val = make_float4(0.f, 0.f, 0.f, 0.f);
            int gr = rowBase0 + row;
            if (gr < nrows) val = ((const float4*)(A + (size_t)gr * DIM))[cv];
            *(float4*)&lds_a[0][row * LDA + (cv << 2)] = val;
        }
    }
    __syncthreads();

    const int wave = tid >> 5;
    const int lane = tid & 31;
    const int half = lane >> 4;   // lane halves own K pairs per the ISA A/B layouts
    const int lm   = lane & 15;
    const int colBase = wave << 4;

    for (int t = 0; t < TILES_PER_BLOCK; ++t) {
        const int rb = rowBase0 + (t << 4);
        if (rb >= nrows) break;  // uniform across the block

        // Stage the NEXT tile into the other buffer (overlaps this tile's math)
        if (t + 1 < TILES_PER_BLOCK) {
            const int nb = rowBase0 + ((t + 1) << 4);
            float* dstbuf = lds_a[(t + 1) & 1];
            for (int i = tid; i < 16 * 32; i += 256) {
                int row = i >> 5;
                int cv  = i & 31;
                float4 val = make_float4(0.f, 0.f, 0.f, 0.f);
                int gr = nb + row;
                if (gr < nrows) val = ((const float4*)(A + (size_t)gr * DIM))[cv];
                *(float4*)&dstbuf[row * LDA + (cv << 2)] = val;
            }
        }

        const float* a_lds = lds_a[t & 1];
        v8f acc = {};
        #pragma unroll
        for (int kk = 0; kk < 32; ++kk) {
            const int k = (kk << 2) + (half << 1);
            // A: lane lm holds row M=lm, K = k, k+1 (16x4 f32 layout)
            float2 av = *(const float2*)&a_lds[lm * LDA + k];
            v2f a; a[0] = av.x; a[1] = av.y;
            // B: lane lm holds col N=colBase+lm, K = k, k+1
            v2f bv;
            bv[0] = lds_w[k * DIM + colBase + lm];
            bv[1] = lds_w[(k + 1) * DIM + colBase + lm];
            acc = __builtin_amdgcn_wmma_f32_16x16x4_f32(
                    false, a, false, bv, (short)0, acc, false, false);
        }

        // D layout: VGPR i -> row = i + 8*half, col = lm
        #pragma unroll
        for (int i = 0; i < 8; ++i) {
            int gr = rb + (half << 3) + i;
            if (gr < nrows) Out[(size_t)gr * DIM + colBase + lm] = acc[i];
        }

        __syncthreads();  // next tile staged & current buffer free for reuse
    }
}

// ------------------------------------------- self-loop init (+bias, added once)
// agg[i,:] = dinv[i]^2 * h[i,:] + bias
__global__ void agg_init_kernel(float* __restrict__ agg,
                                const float* __restrict__ h,
                                const float* __restrict__ dinv,
                                const float* __restrict__ bias, int n) {
    int i = blockIdx.x * blockDim.x + threadIdx.x;  // over n * 32 float4s
    int total = n * (DIM / 4);
    if (i < total) {
        int node = i >> 5;
        int dv   = (i & 31) << 2;
        float s = dinv[node]; s = s * s;
        float4 hv = ((const float4*)h)[i];
        float4 o;
        o.x = s * hv.x + bias[dv + 0];
        o.y = s * hv.y + bias[dv + 1];
        o.z = s * hv.z + bias[dv + 2];
        o.w = s * hv.w + bias[dv + 3];
        ((float4*)agg)[i] = o;
    }
}

// ------------------------------------------------- edge scatter: wave per edge
// Working set (h + agg ~ 102 MB) is L2-resident on MI455X's 192 MB L2:
// float4 gather of the source row + 4 atomic f32 adds per lane.
__global__ __launch_bounds__(256) void agg_edges_kernel(
        float* __restrict__ agg, const float* __restrict__ h,
        const long long* __restrict__ ei, const float* __restrict__ norm, int E) {
    int wid  = blockIdx.x * (blockDim.x >> 5) + (threadIdx.x >> 5);
    int lane = threadIdx.x & 31;
    if (wid >= E) return;
    long long r = ei[wid];
    long long c = ei[(long long)E + wid];
    float nm = norm[wid];
    float4 v = ((const float4*)(h + r * (long long)DIM))[lane];
    float* dst = agg + c * (long long)DIM + (lane << 2);
    atomicAdd(dst + 0, nm * v.x);
    atomicAdd(dst + 1, nm * v.y);
    atomicAdd(dst + 2, nm * v.z);
    atomicAdd(dst + 3, nm * v.w);
}

// ------------------------------------------------------------------ ReLU
__global__ void relu_kernel(float* __restrict__ dst,
                            const float* __restrict__ src, int total4) {
    int i = blockIdx.x * blockDim.x + threadIdx.x;
    if (i < total4) {
        float4 v = ((const float4*)src)[i];
        v.x = fmaxf(v.x, 0.f); v.y = fmaxf(v.y, 0.f);
        v.z = fmaxf(v.z, 0.f); v.w = fmaxf(v.w, 0.f);
        ((float4*)dst)[i] = v;
    }
}

// ------------------------------------------------------------------ launch
extern "C" void kernel_launch(void* const* d_in, const int* in_sizes, int n_in,
                              void* d_out, int out_size, void* d_ws, size_t ws_size,
                              hipStream_t stream) {
    const float*     x  = (const float*)d_in[0];
    const long long* ei = (const long long*)d_in[1];  // [2, E] int64
    const float*     ew = (const float*)d_in[2];
    const float*     W1 = (const float*)d_in[3];
    const float*     b1 = (const float*)d_in[4];
    const float*     W2 = (const float*)d_in[5];
    const float*     b2 = (const float*)d_in[6];
    float* out = (float*)d_out;

    const int N = in_sizes[0] / DIM;
    const int E = in_sizes[2];

    // workspace layout (floats), offsets rounded to 256 elements
    float* ws = (float*)d_ws;
    size_t off = 0;
    float* dinv = ws + off; off += (size_t)N;        off = (off + 255) & ~(size_t)255;
    float* norm = ws + off; off += (size_t)E;        off = (off + 255) & ~(size_t)255;
    float* hbuf = ws + off; off += (size_t)N * DIM;  off = (off + 255) & ~(size_t)255;
    float* aggbuf = ws + off;

    const int T = 256;
    const int nBlkN  = (N + T - 1) / T;
    const int nBlkE  = (E + T - 1) / T;
    const int nBlkV  = (N * (DIM / 4) + T - 1) / T;               // elementwise float4
    const int nBlkEw = (E + 7) / 8;                               // 8 edges (waves)/block
    const int nBlkG  = (N + 16 * TILES_PER_BLOCK - 1) / (16 * TILES_PER_BLOCK);

    // normalization
    init_deg_kernel <<<nBlkN, T, 0, stream>>>(dinv, N);
    deg_accum_kernel<<<nBlkE, T, 0, stream>>>(dinv, ei, ew, E);
    dinv_kernel     <<<nBlkN, T, 0, stream>>>(dinv, N);
    norm_kernel     <<<nBlkE, T, 0, stream>>>(norm, dinv, ei, ew, E);

    // layer 1: h = x@W1 ; agg = selfloop + b1 + scatter ; h1 = relu(agg)
    gemm128_wmma_kernel<<<nBlkG, T, 0, stream>>>(x, W1, hbuf, N);
    agg_init_kernel    <<<nBlkV, T, 0, stream>>>(aggbuf, hbuf, dinv, b1, N);
    agg_edges_kernel   <<<nBlkEw, T, 0, stream>>>(aggbuf, hbuf, ei, norm, E);
    relu_kernel        <<<nBlkV, T, 0, stream>>>(hbuf, aggbuf, N * (DIM / 4));

    // layer 2: h2 = h1@W2 ; out = selfloop + b2 + scatter
    gemm128_wmma_kernel<<<nBlkG, T, 0, stream>>>(hbuf, W2, aggbuf, N);
    agg_init_kernel    <<<nBlkV, T, 0, stream>>>(out, aggbuf, dinv, b2, N);
    agg_edges_kernel   <<<nBlkEw, T, 0, stream>>>(out, aggbuf, ei, norm, E);
}